// DynamicMaskHead_76390288326983
// MI455X (gfx1250) — compile-verified
//
#include <hip/hip_runtime.h>
#include <hip/hip_bf16.h>

typedef float v2f __attribute__((ext_vector_type(2)));
typedef float v8f __attribute__((ext_vector_type(8)));

// Fast sigmoid/silu: v_exp_f32 + v_rcp_f32 (1-ulp), avoids the ~10-op
// IEEE division expansion (v_div_scale/v_div_fmas chain).
__device__ __forceinline__ float sigmoid_f(float x) {
    return __builtin_amdgcn_rcpf(1.0f + __expf(-x));
}
__device__ __forceinline__ float silu_f(float v) {
    return v * sigmoid_f(v);
}

// ---------------------------------------------------------------------------
// Stage 1: dynamic-weight MLP -> logits (N x 96 x 96)
// One wave per 16-pixel tile; V_WMMA_F32_16X16X4_F32 for layers 0 and 1.
// Zero-padding is branchless (index/result selects -> v_cndmask feeding the
// WMMAs); per-instance scalars forced wave-uniform via readfirstlane so
// metadata + bias/w2 loads lower to s_load.
// ---------------------------------------------------------------------------
__global__ void __launch_bounds__(256)
dmh_mlp_wmma_kernel(const float* __restrict__ mask_feats,   // (B,8,96,96)
                    const float* __restrict__ pc,           // (N,169)
                    const float* __restrict__ pos,          // (N,2) (y,x)
                    const float* __restrict__ soi,          // (3,)
                    const int*   __restrict__ img_inds,     // (N,)
                    const int*   __restrict__ levels,       // (N,)
                    const int*   __restrict__ stride_p,     // scalar (=8)
                    float*       __restrict__ logits,       // (N,9216)
                    int N)
{
    const int wave = blockIdx.x * (blockDim.x >> 5) + (threadIdx.x >> 5);
    const int lane = threadIdx.x & 31;
    if (wave >= N * 576) return;                 // wave-uniform guard

    // n / tile are wave-uniform: tell the compiler so metadata goes scalar.
    const int n    = __builtin_amdgcn_readfirstlane(wave / 576);
    const int tile = __builtin_amdgcn_readfirstlane(wave % 576);
    const int base = tile * 16;

    const float stride  = (float)stride_p[0];
    const int   img     = img_inds[n];
    const float inv_soi = 1.0f / soi[levels[n]];
    const float pos_y   = pos[n * 2 + 0];
    const float pos_x   = pos[n * 2 + 1];
    const float* w      = pc + (size_t)n * 169;

    const int row = lane & 15;   // A: out-channel row / B: pixel column
    const int hi  = lane >> 4;   // lane half
    const int kb  = hi * 2;      // K sub-offset within a K=4 group

    // ---- A0: W0 (8x10) zero-padded to 16x12. Indices row*10+k <= 161 are
    // always in-range of the 169-float block: unconditional load + cndmask.
    v2f a0[3];
#pragma unroll
    for (int g = 0; g < 3; ++g) {
        const int k0 = 4 * g + kb;
        const float vx = w[row * 10 + k0];
        const float vy = w[row * 10 + k0 + 1];
        a0[g].x = (row < 8 && k0     < 10) ? vx : 0.0f;
        a0[g].y = (row < 8 && k0 + 1 < 10) ? vy : 0.0f;
    }

    // ---- B0: X (12x16): rows {rel_y, rel_x, feat0..7, 0, 0}, col = pixel ----
    const int   p     = base + row;
    const int   py    = p / 96, px = p % 96;
    const float rel_y = (pos_y - ((float)py * stride + 0.5f * stride)) * inv_soi;
    const float rel_x = (pos_x - ((float)px * stride + 0.5f * stride)) * inv_soi;
    const float* f    = mask_feats + (size_t)img * (8 * 9216) + p;

    // Branchless row fetch: clamp channel into [0,7] for an always-valid load,
    // then patch in rel rows / zero rows with selects.
    auto xrow = [&](int r) -> float {
        int ch = r - 2;
        int chc = ch < 0 ? 0 : (ch > 7 ? 7 : ch);
        float fv = f[chc * 9216];
        float val = (r >= 2 && r < 10) ? fv : 0.0f;
        val = (r == 0) ? rel_y : val;
        val = (r == 1) ? rel_x : val;
        return val;
    };

    v2f b0m[3];
#pragma unroll
    for (int g = 0; g < 3; ++g) {
        const int r0 = 4 * g + kb;
        b0m[g].x = xrow(r0);
        b0m[g].y = xrow(r0 + 1);
    }

    // ---- Layer 0: 3 chained f32 WMMAs (K=4 each) ----
    v8f acc0 = {};
#pragma unroll
    for (int g = 0; g < 3; ++g)
        acc0 = __builtin_amdgcn_wmma_f32_16x16x4_f32(false, a0[g], false, b0m[g],
                                                     (short)0, acc0, false, false);

    // bias (offset 152, scalar loads) + silu; valid channels in lanes 0..15
    float h[8];
#pragma unroll
    for (int r = 0; r < 8; ++r)
        h[r] = silu_f(acc0[r] + w[152 + r]);

    // ---- Repack h (channel-in-VGPR, pixel-in-lane) into B layout (shuffles)
    v2f b1m[2];
#pragma unroll
    for (int g = 0; g < 2; ++g) {
        const float v0 = __shfl(h[4 * g + 0], row);
        const float v1 = __shfl(h[4 * g + 1], row);
        const float v2 = __shfl(h[4 * g + 2], row);
        const float v3 = __shfl(h[4 * g + 3], row);
        b1m[g].x = hi ? v2 : v0;
        b1m[g].y = hi ? v3 : v1;
    }

    // ---- A1: W1 (8x8) at offset 80, zero-padded to 16x8. Index could exceed
    // the 169-float block for row>=8, so select the index (stays branchless).
    v2f a1[2];
#pragma unroll
    for (int g = 0; g < 2; ++g) {
        const int k0 = 4 * g + kb;
        const int ia = (row < 8) ? (80 + row * 8 + k0) : 0;
        const float vx = w[ia];
        const float vy = w[ia + 1];
        a1[g].x = (row < 8) ? vx : 0.0f;
        a1[g].y = (row < 8) ? vy : 0.0f;
    }

    v8f acc1 = {};
    acc1 = __builtin_amdgcn_wmma_f32_16x16x4_f32(false, a1[0], false, b1m[0],
                                                 (short)0, acc1, false, false);
    acc1 = __builtin_amdgcn_wmma_f32_16x16x4_f32(false, a1[1], false, b1m[1],
                                                 (short)0, acc1, false, false);

    // ---- Layer 2: 1x8 dot (VALU); w2@144, b1@160, b2@168 (scalar loads) ----
    float out = w[168];
#pragma unroll
    for (int r = 0; r < 8; ++r)
        out += w[144 + r] * silu_f(acc1[r] + w[160 + r]);

    if (lane < 16)
        logits[(size_t)n * 9216 + p] = out;
}

// ---------------------------------------------------------------------------
// Stage 2: sigmoid(aligned_bilinear(logits, 2)) -> sig (N x 192 x 192)
// U[y][x] = G[max(y-1,0)][max(x-1,0)], G = half-step bilinear on edge-padded L
// ---------------------------------------------------------------------------
__global__ void __launch_bounds__(256)
dmh_upsig_kernel(const float* __restrict__ logits, float* __restrict__ sig, int N)
{
    const int idx = blockIdx.x * blockDim.x + threadIdx.x;
    if (idx >= N * 192 * 192) return;

    const int x = idx % 192;
    const int y = (idx / 192) % 192;
    const int n = idx / (192 * 192);
    const float* L = logits + (size_t)n * 9216;

    const int yy = (y > 0) ? y - 1 : 0;
    const int xx = (x > 0) ? x - 1 : 0;
    const int y0 = yy >> 1;
    const int x0 = xx >> 1;
    const float fy = (yy & 1) ? 0.5f : 0.0f;
    const float fx = (xx & 1) ? 0.5f : 0.0f;
    const int y1 = (y0 + 1 > 95) ? 95 : y0 + 1;  // edge-pad replication
    const int x1 = (x0 + 1 > 95) ? 95 : x0 + 1;

    const float v00 = L[y0 * 96 + x0];
    const float v01 = L[y0 * 96 + x1];
    const float v10 = L[y1 * 96 + x0];
    const float v11 = L[y1 * 96 + x1];
    const float top = v00 + fx * (v01 - v00);
    const float bot = v10 + fx * (v11 - v10);
    const float u   = top + fy * (bot - top);

    sig[idx] = sigmoid_f(u);
}

// ---------------------------------------------------------------------------
// Stage 3: _resize_ac 192 -> 768 (both axes), float4 stores (write-BW bound)
// ---------------------------------------------------------------------------
__global__ void __launch_bounds__(256)
dmh_resize_out_kernel(const float* __restrict__ sig, float* __restrict__ out, int N)
{
    const int idx = blockIdx.x * blockDim.x + threadIdx.x;
    if (idx >= N * 768 * 192) return;

    const int xq = idx % 192;           // x quad (4 output cols)
    const int oy = (idx / 192) % 768;
    const int n  = idx / (192 * 768);
    const float* s = sig + (size_t)n * 192 * 192;

    const float scale = 191.0f / 767.0f;
    const float pyf = (float)oy * scale;
    int y0 = (int)pyf; if (y0 > 190) y0 = 190;
    const float fy = pyf - (float)y0;
    const float* r0 = s + y0 * 192;
    const float* r1 = r0 + 192;

    float4 res;
    float* rp = (float*)&res;
#pragma unroll
    for (int j = 0; j < 4; ++j) {
        const int ox = xq * 4 + j;
        const float pxf = (float)ox * scale;
        int x0 = (int)pxf; if (x0 > 190) x0 = 190;
        const float fx = pxf - (float)x0;
        const float a0 = r0[x0], a1 = r0[x0 + 1];
        const float b0 = r1[x0], b1 = r1[x0 + 1];
        const float top = a0 + fx * (a1 - a0);
        const float bot = b0 + fx * (b1 - b0);
        rp[j] = top + fy * (bot - top);
    }
    ((float4*)out)[idx] = res;
}

// ---------------------------------------------------------------------------
extern "C" void kernel_launch(void* const* d_in, const int* in_sizes, int n_in,
                              void* d_out, int out_size, void* d_ws, size_t ws_size,
                              hipStream_t stream)
{
    (void)n_in; (void)out_size; (void)ws_size;

    const float* mask_feats = (const float*)d_in[0];
    const float* pc         = (const float*)d_in[1];
    const float* pos        = (const float*)d_in[2];
    const float* soi        = (const float*)d_in[3];
    const int*   img_inds   = (const int*)d_in[4];
    const int*   levels     = (const int*)d_in[5];
    const int*   stride_p   = (const int*)d_in[6];

    const int N = in_sizes[4];          // img_inds length

    float* logits = (float*)d_ws;                       // N * 9216 floats
    float* sig    = logits + (size_t)N * 9216;          // N * 192*192 floats
    float* out    = (float*)d_out;                      // N * 768*768 floats

    {   // stage 1: 8 waves / block, one 16-pixel tile per wave
        const int waves  = N * 576;
        const int blocks = (waves + 7) / 8;
        dmh_mlp_wmma_kernel<<<blocks, 256, 0, stream>>>(
            mask_feats, pc, pos, soi, img_inds, levels, stride_p, logits, N);
    }
    {   // stage 2
        const int total  = N * 192 * 192;
        const int blocks = (total + 255) / 256;
        dmh_upsig_kernel<<<blocks, 256, 0, stream>>>(logits, sig, N);
    }
    {   // stage 3
        const int total  = N * 768 * 192;
        const int blocks = (total + 255) / 256;
        dmh_resize_out_kernel<<<blocks, 256, 0, stream>>>(sig, out, N);
    }
}